// MoChA_3375844295004
// MI455X (gfx1250) — compile-verified
//
#include <hip/hip_runtime.h>
#include <hip/hip_bf16.h>

// ---------------------------------------------------------------------------
// MoChA fused attention for MI455X (gfx1250, wave32).
// Roofline: key(64MB)+value(64MB) streamed once -> ~5.5us floor @ 23.3 TB/s.
// FLOPs (~0.5G incl. tanh) are far below compute ceilings -> bandwidth bound.
// f32 WMMA (V_WMMA_F32_16X16X4_F32) used for the dense q = dec_h @ W.T GEMMs.
// ---------------------------------------------------------------------------

typedef __attribute__((ext_vector_type(2))) float v2f;
typedef __attribute__((ext_vector_type(8))) float v8f;

constexpr int kB    = 32;    // batch
constexpr int kSeq  = 1024;  // S
constexpr int kAtt  = 512;   // attention dim (== decoder dim here)
constexpr int kD    = 512;   // enc/value dim
constexpr int kW    = 8;     // MoChA chunk
constexpr int kSCH  = 16;    // S-chunks for context partial reduction
constexpr float kLog2e = 1.4426950408889634f;

__device__ __forceinline__ float fexp(float x)  { return __builtin_amdgcn_exp2f(x * kLog2e); }
__device__ __forceinline__ float fsigmoid(float x){
    return __builtin_amdgcn_rcpf(1.0f + fexp(-x));
}
__device__ __forceinline__ float ftanh(float x) {
    // tanh(x) = 1 - 2/(exp(2x)+1); v_exp_f32 + v_rcp_f32, saturates correctly.
    float e = __builtin_amdgcn_exp2f(x * (2.0f * kLog2e));
    return 1.0f - 2.0f * __builtin_amdgcn_rcpf(e + 1.0f);
}

// ---------------------------------------------------------------------------
// Kernel 1: weight-normed v vectors:  vn = g * v / ||v||
// ---------------------------------------------------------------------------
__global__ void vnorm_kernel(const float* __restrict__ vm, const float* __restrict__ gm,
                             const float* __restrict__ vc, const float* __restrict__ gc,
                             float* __restrict__ vmn, float* __restrict__ vcn) {
    __shared__ float red[kAtt];
    const int tid = threadIdx.x;
    const float* v = blockIdx.x ? vc : vm;
    const float* g = blockIdx.x ? gc : gm;
    float* o       = blockIdx.x ? vcn : vmn;
    float x = v[tid];
    red[tid] = x * x;
    __syncthreads();
    for (int off = kAtt / 2; off > 0; off >>= 1) {
        if (tid < off) red[tid] += red[tid + off];
        __syncthreads();
    }
    o[tid] = g[0] * x * rsqrtf(red[0]);
}

// ---------------------------------------------------------------------------
// Kernel 2: q = dec_h @ W.T + bias via f32 WMMA 16x16x4.
// One wave per 16x16 output tile. grid = (Ntiles=32, Mtiles=2, mat=2).
// A (16x4 MxK): lane<16 holds A[lane, k0..k0+1], lane>=16 holds A[lane-16, 2..3]
// B (4x16 KxN): lane<16 holds B[k0..k0+1, lane], lane>=16 holds B[2..3, lane-16]
// C (16x16):    vgpr r, lane<16 -> (M=r, N=lane); lane>=16 -> (M=r+8, N=lane-16)
// ---------------------------------------------------------------------------
__global__ void qgemm_wmma(const float* __restrict__ dh,
                           const float* __restrict__ Wm, const float* __restrict__ bm,
                           const float* __restrict__ Wc, const float* __restrict__ bc,
                           float* __restrict__ qm, float* __restrict__ qc) {
    const int lane  = threadIdx.x & 31;
    const int ntile = blockIdx.x;   // 0..31  (ATT/16)
    const int mtile = blockIdx.y;   // 0..1   (B/16)
    const int mat   = blockIdx.z;   // 0 = mono, 1 = chunk
    const float* W    = mat ? Wc : Wm;
    const float* bias = mat ? bc : bm;
    float* out        = mat ? qc : qm;

    const int mrow = mtile * 16 + (lane & 15);
    const int nrow = ntile * 16 + (lane & 15);
    const int klo  = (lane >> 4) * 2;

    v8f c = {0.f, 0.f, 0.f, 0.f, 0.f, 0.f, 0.f, 0.f};

#if __has_builtin(__builtin_amdgcn_wmma_f32_16x16x4_f32)
    const float* arow = dh + mrow * kAtt;  // dec_h row (K = 512 decoder dim)
    const float* brow = W  + nrow * kAtt;  // W row (q[m,n] = sum_k dh[m,k]*W[n,k])
    for (int kk = 0; kk < kAtt; kk += 4) {
        v2f a, b;
        v2f av = *(const v2f*)(arow + kk + klo);
        v2f bv = *(const v2f*)(brow + kk + klo);
        a = av; b = bv;
        c = __builtin_amdgcn_wmma_f32_16x16x4_f32(false, a, false, b,
                                                  (short)0, c, false, false);
    }
#else
    // Scalar fallback producing identical C layout.
    for (int r = 0; r < 8; ++r) {
        int mm = mtile * 16 + r + ((lane >> 4) << 3);
        int nn = ntile * 16 + (lane & 15);
        float acc = 0.f;
        for (int k = 0; k < kAtt; ++k) acc += dh[mm * kAtt + k] * W[nn * kAtt + k];
        c[r] = acc;
    }
#endif

    const int nn    = ntile * 16 + (lane & 15);
    const int mbase = mtile * 16 + ((lane >> 4) << 3);
    const float bv  = bias[nn];
#pragma unroll
    for (int r = 0; r < 8; ++r)
        out[(mbase + r) * kAtt + nn] = c[r] + bv;
}

// ---------------------------------------------------------------------------
// Kernel 3: fused dual energies. One wave per (b,s) row; key read ONCE for
// both mono and chunk energies. 4 x float4 coalesced loads per lane.
//   e[b,s] = sum_a vn[a]*tanh(q[b,a]+key[b,s,a]) + vbias + r
// ---------------------------------------------------------------------------
__global__ void energy_kernel(const float* __restrict__ key,
                              const float* __restrict__ qm, const float* __restrict__ qc,
                              const float* __restrict__ vmn, const float* __restrict__ vcn,
                              const float* __restrict__ vbm, const float* __restrict__ rm,
                              const float* __restrict__ vbc, const float* __restrict__ rc,
                              float* __restrict__ me, float* __restrict__ ce) {
    const int wave = threadIdx.x >> 5;
    const int lane = threadIdx.x & 31;
    const int row  = blockIdx.x * 8 + wave;        // flat b*S+s, 0..32767
    const int b    = row >> 10;
    const float* krow = key + (size_t)row * kAtt;
    const float* qmr  = qm + b * kAtt;
    const float* qcr  = qc + b * kAtt;

    float accm = 0.f, accc = 0.f;
#pragma unroll
    for (int i = 0; i < 4; ++i) {
        const int a = i * 128 + lane * 4;
        float4 k4  = *(const float4*)(krow + a);
        float4 qm4 = *(const float4*)(qmr + a);
        float4 qc4 = *(const float4*)(qcr + a);
        float4 vm4 = *(const float4*)(vmn + a);
        float4 vc4 = *(const float4*)(vcn + a);
        accm += vm4.x * ftanh(qm4.x + k4.x);
        accm += vm4.y * ftanh(qm4.y + k4.y);
        accm += vm4.z * ftanh(qm4.z + k4.z);
        accm += vm4.w * ftanh(qm4.w + k4.w);
        accc += vc4.x * ftanh(qc4.x + k4.x);
        accc += vc4.y * ftanh(qc4.y + k4.y);
        accc += vc4.z * ftanh(qc4.z + k4.z);
        accc += vc4.w * ftanh(qc4.w + k4.w);
    }
#pragma unroll
    for (int off = 16; off > 0; off >>= 1) {
        accm += __shfl_xor(accm, off, 32);
        accc += __shfl_xor(accc, off, 32);
    }
    if (lane == 0) {
        me[row] = accm + vbm[0] + rm[0];
        ce[row] = accc + vbc[0] + rc[0];
    }
}

// ---------------------------------------------------------------------------
// Kernel 4: per-batch scans. One block of 1024 threads (= S) per batch row.
// Hillis-Steele scans in LDS (10 steps) + direct 8-wide MoChA windows.
// ---------------------------------------------------------------------------
template <bool MUL>
__device__ __forceinline__ float inclusive_scan(float* buf, int s, float v) {
    __syncthreads();          // protect previous readers of buf
    buf[s] = v;
    __syncthreads();
    for (int off = 1; off < kSeq; off <<= 1) {
        float t = buf[s];
        float o = (s >= off) ? buf[s - off] : (MUL ? 1.0f : 0.0f);
        __syncthreads();
        buf[s] = MUL ? (t * o) : (t + o);
        __syncthreads();
    }
    return buf[s];
}

__global__ void __launch_bounds__(1024)
mocha_scan(const float* __restrict__ me, const float* __restrict__ cein,
           const float* __restrict__ prev_att, const float* __restrict__ noise,
           float* __restrict__ beta_out) {
    __shared__ float sbuf[kSeq];
    __shared__ float cebuf[kSeq];
    __shared__ float eubuf[kSeq];
    __shared__ float t2buf[kSeq];
    const int b = blockIdx.x, s = threadIdx.x;
    const int idx = b * kSeq + s;

    cebuf[s] = cein[idx];                       // stash chunk energy row

    float p = fsigmoid(me[idx] + noise[idx]);
    if (s == kSeq - 1) p = 1.0f;                // end-mask
    float prev = prev_att[idx];

    // normalize prev attention
    inclusive_scan<false>(sbuf, s, prev);
    float total = sbuf[kSeq - 1];
    float prevn = prev / total;

    // exclusive cumprod of (1-p)
    inclusive_scan<true>(sbuf, s, 1.0f - p);
    float cp = (s == 0) ? 1.0f : sbuf[s - 1];
    float cpc = fmaxf(cp, 1e-20f);              // cp in [0,1] -> only lower clip

    // alpha = p * cp * cumsum(prevn / cpc)
    float csum = inclusive_scan<false>(sbuf, s, prevn / cpc);
    float alpha = p * cp * csum;

    // --- MoChA stable chunkwise attention, w = 8 ---
    __syncthreads();
    float cval = cebuf[s];
    float mm = cval;
#pragma unroll
    for (int j = 1; j < kW; ++j) { int k = s - j; if (k >= 0) mm = fmaxf(mm, cebuf[k]); }
    float eu = fexp(cval - mm);
    eubuf[s] = eu;
    __syncthreads();
    float den = 0.f;
#pragma unroll
    for (int j = 0; j < kW; ++j) { int k = s - j; if (k >= 0) den += eubuf[k]; }
    den = fmaxf(den, 1e-10f);
    t2buf[s] = alpha / den;
    __syncthreads();
    float ms = 0.f;
#pragma unroll
    for (int j = 0; j < kW; ++j) { int k = s + j; if (k < kSeq) ms += t2buf[k]; }
    beta_out[idx] = eu * ms;
}

// ---------------------------------------------------------------------------
// Kernel 5: context partials. grid=(kSCH, B), 256 thr; each thread owns 2 d's
// (float2 coalesced), reduces 64 s-rows -> deterministic (no atomics).
// ---------------------------------------------------------------------------
__global__ void context_partial(const float* __restrict__ beta,
                                const float* __restrict__ value,
                                float* __restrict__ partial) {
    const int chunk = blockIdx.x;         // 0..kSCH-1
    const int b     = blockIdx.y;
    const int tid   = threadIdx.x;        // 0..255
    const int d     = tid * 2;
    const int s0    = chunk * (kSeq / kSCH);
    float2 acc = make_float2(0.f, 0.f);
    for (int s = s0; s < s0 + kSeq / kSCH; ++s) {
        float bt = beta[b * kSeq + s];
        float2 v = *(const float2*)(value + ((size_t)(b * kSeq + s)) * kD + d);
        acc.x += bt * v.x;
        acc.y += bt * v.y;
    }
    *(float2*)(partial + ((size_t)chunk * kB + b) * kD + d) = acc;
}

__global__ void context_reduce(const float* __restrict__ partial,
                               float* __restrict__ out) {
    const int b = blockIdx.x, d = threadIdx.x;  // 512 threads
    float acc = 0.f;
#pragma unroll
    for (int c = 0; c < kSCH; ++c)
        acc += partial[((size_t)c * kB + b) * kD + d];
    out[b * kD + d] = acc;
}

// ---------------------------------------------------------------------------
extern "C" void kernel_launch(void* const* d_in, const int* in_sizes, int n_in,
                              void* d_out, int out_size, void* d_ws, size_t ws_size,
                              hipStream_t stream) {
    const float* dh    = (const float*)d_in[0];
    const float* key   = (const float*)d_in[1];
    const float* value = (const float*)d_in[2];
    const float* prev  = (const float*)d_in[3];
    const float* noise = (const float*)d_in[4];
    const float* Wm  = (const float*)d_in[5];
    const float* bm  = (const float*)d_in[6];
    const float* vm  = (const float*)d_in[7];
    const float* gm  = (const float*)d_in[8];
    const float* vbm = (const float*)d_in[9];
    const float* rm  = (const float*)d_in[10];
    const float* Wc  = (const float*)d_in[11];
    const float* bc  = (const float*)d_in[12];
    const float* vc  = (const float*)d_in[13];
    const float* gc  = (const float*)d_in[14];
    const float* vbc = (const float*)d_in[15];
    const float* rc  = (const float*)d_in[16];
    float* out = (float*)d_out;

    // workspace layout (floats): ~1.54 MB total
    float* ws   = (float*)d_ws;
    float* qm   = ws;                    // [32,512]
    float* qc   = qm  + kB * kAtt;       // [32,512]
    float* vmn  = qc  + kB * kAtt;       // [512]
    float* vcn  = vmn + kAtt;            // [512]
    float* me   = vcn + kAtt;            // [32,1024]
    float* ce   = me  + kB * kSeq;       // [32,1024]
    float* beta = ce  + kB * kSeq;       // [32,1024]
    float* part = beta + kB * kSeq;      // [kSCH,32,512]

    vnorm_kernel<<<dim3(2), dim3(kAtt), 0, stream>>>(vm, gm, vc, gc, vmn, vcn);
    qgemm_wmma<<<dim3(kAtt / 16, kB / 16, 2), dim3(32), 0, stream>>>(
        dh, Wm, bm, Wc, bc, qm, qc);
    energy_kernel<<<dim3((kB * kSeq) / 8), dim3(256), 0, stream>>>(
        key, qm, qc, vmn, vcn, vbm, rm, vbc, rc, me, ce);
    mocha_scan<<<dim3(kB), dim3(kSeq), 0, stream>>>(me, ce, prev, noise, beta);
    context_partial<<<dim3(kSCH, kB), dim3(256), 0, stream>>>(beta, value, part);
    context_reduce<<<dim3(kB), dim3(kD), 0, stream>>>(part, out);
}